// AcyclicREN_72602127172158
// MI455X (gfx1250) — compile-verified
//
#include <hip/hip_runtime.h>
#include <hip/hip_bf16.h>
#include <cmath>

// ---------------- CDNA5 fp32 WMMA plumbing ----------------
typedef float f32x2 __attribute__((ext_vector_type(2)));
typedef float f32x8 __attribute__((ext_vector_type(8)));

#define EPS_F 1e-3f

// D = A(16x4) * B(4x16) + C, fp32 throughout (V_WMMA_F32_16X16X4_F32).
__device__ __forceinline__ f32x8 wmma4(f32x2 a, f32x2 b, f32x8 c) {
  return __builtin_amdgcn_wmma_f32_16x16x4_f32(
      /*neg_a=*/false, a, /*neg_b=*/false, b,
      /*c_mod=*/(short)0, c, /*reuse_a=*/false, /*reuse_b=*/false);
}

// Lane mapping (ISA 7.12.2, wave32):
//  A frag : lane holds A[m, k2+0..1], m = lane&15, k2 = 2*(lane>>4)
//  B frag : lane holds B[k2+0..1, n], n = lane&15
//  C/D    : vgpr v holds D[cm+v, cn], cn = lane&15, cm = 8*(lane>>4)

// ---------------- Generic one-16x16-tile-per-wave WMMA GEMM ----------------
// out[M,N] = op(A)[M,K] * B[K,N] (+ Cin) (+ diag_add on diagonal)
// AT: A stored as A[k*lda + m]  (math-A is the transpose of storage)
// BT: B stored as B[n*ldb + k]  (storage is W with math-B = W^T)
// Software-pipelined: next k-step fragments are issued before current WMMA.
template <bool AT, bool BT>
__global__ void __launch_bounds__(256) k_gemm(
    float* __restrict__ out, int ldo,
    const float* __restrict__ A, int lda,
    const float* __restrict__ B, int ldb,
    const float* __restrict__ Cin, int ldc,
    int M, int N, int K, float diag_add, int skip_r0, int skip_r1) {
  int lane = threadIdx.x & 31;
  int wave = (blockIdx.x * blockDim.x + threadIdx.x) >> 5;
  int tilesN = N >> 4;
  int tiles = (M >> 4) * tilesN;
  if (wave >= tiles) return;
  int m0 = (wave / tilesN) << 4;
  int n0 = (wave % tilesN) << 4;
  if (m0 >= skip_r0 && m0 < skip_r1) return;  // skip pivot rows (elimination)
  int lm = lane & 15;
  int lk = (lane >> 4) << 1;
  int cn = lane & 15;
  int cm = (lane >> 4) << 3;

  f32x8 c;
#pragma unroll
  for (int v = 0; v < 8; ++v)
    c[v] = Cin ? Cin[(size_t)(m0 + cm + v) * ldc + n0 + cn] : 0.0f;

  auto loadA = [&](int k) -> f32x2 {
    f32x2 a;
    if (AT) {
      a.x = A[(size_t)(k + lk) * lda + m0 + lm];
      a.y = A[(size_t)(k + lk + 1) * lda + m0 + lm];
    } else {
      a = *(const f32x2*)(A + (size_t)(m0 + lm) * lda + k + lk);
    }
    return a;
  };
  auto loadB = [&](int k) -> f32x2 {
    f32x2 b;
    if (BT) {
      b = *(const f32x2*)(B + (size_t)(n0 + lm) * ldb + k + lk);
    } else {
      b.x = B[(size_t)(k + lk) * ldb + n0 + lm];
      b.y = B[(size_t)(k + lk + 1) * ldb + n0 + lm];
    }
    return b;
  };

  f32x2 a_c = loadA(0), b_c = loadB(0), a_n, b_n;
  for (int k = 0; k < K; k += 4) {
    bool hn = (k + 4) < K;
    if (hn) { a_n = loadA(k + 4); b_n = loadB(k + 4); }
    c = wmma4(a_c, b_c, c);
    if (hn) { a_c = a_n; b_c = b_n; }
  }
#pragma unroll
  for (int v = 0; v < 8; ++v) {
    int r = m0 + cm + v, col = n0 + cn;
    float val = c[v];
    if (r == col) val += diag_add;
    out[(size_t)r * ldo + col] = val;
  }
}

// ---------------- derived-parameter kernels ----------------
// Build Mgj = [E | I] (512x1024) and D11 = -tril(H22,-1)
__global__ void k_build_EM(const float* __restrict__ H, const float* __restrict__ Y,
                           float* __restrict__ Mgj, float* __restrict__ D11) {
  int idx = blockIdx.x * blockDim.x + threadIdx.x;
  int i = idx >> 9, j = idx & 511;
  if (i >= 512) return;
  float e = 0.5f * (H[(size_t)i * 1536 + j] + H[(size_t)(1024 + i) * 1536 + 1024 + j] +
                    Y[(size_t)i * 512 + j] - Y[(size_t)j * 512 + i]);
  Mgj[(size_t)i * 1024 + j] = e;
  Mgj[(size_t)i * 1024 + 512 + j] = (i == j) ? 1.0f : 0.0f;
  D11[(size_t)i * 512 + j] = (j < i) ? -H[(size_t)(512 + i) * 1536 + 512 + j] : 0.0f;
}

// rlam = 1/Lam = 2/diag(H22); bpre = C1 @ x0 = -H21 @ x0; bz = Fm @ x0
__global__ void k_vecs(const float* __restrict__ H, const float* __restrict__ x0,
                       float* __restrict__ rlam, float* __restrict__ bpre,
                       float* __restrict__ bz) {
  int i = blockIdx.x * blockDim.x + threadIdx.x;
  if (i >= 512) return;
  rlam[i] = 2.0f / H[(size_t)(512 + i) * 1536 + 512 + i];
  float s1 = 0.f, s2 = 0.f;
  for (int j = 0; j < 512; ++j) {
    float xj = x0[j];
    s1 += H[(size_t)(512 + i) * 1536 + j] * xj;
    s2 += H[(size_t)(1024 + i) * 1536 + j] * xj;
  }
  bpre[i] = -s1;
  bz[i] = s2;
}

// bias_y = G @ bz
__global__ void k_biasy(const float* __restrict__ G, const float* __restrict__ bz,
                        float* __restrict__ biasy) {
  int o = blockIdx.x * blockDim.x + threadIdx.x;
  if (o >= 256) return;
  float s = 0.f;
  for (int t = 0; t < 512; ++t) s += G[(size_t)o * 512 + t] * bz[t];
  biasy[o] = s;
}

// ---------------- blocked Gauss-Jordan helpers ----------------
// Invert the current 64x64 diagonal block in LDS (one workgroup, unpivoted; E has
// PD symmetric part so pivots are safe).
__global__ void k_gj_diag(const float* __restrict__ Mgj, float* __restrict__ invD, int pk) {
  __shared__ float s[64][129];
  __shared__ float colbuf[64];
  int tid = threadIdx.x;  // 256
  for (int idx = tid; idx < 64 * 64; idx += 256) {
    int r = idx >> 6, c = idx & 63;
    s[r][c] = Mgj[(size_t)(pk + r) * 1024 + pk + c];
    s[r][64 + c] = (r == c) ? 1.0f : 0.0f;
  }
  __syncthreads();
  for (int p = 0; p < 64; ++p) {
    if (tid < 64) colbuf[tid] = s[tid][p];
    __syncthreads();
    float pinv = 1.0f / colbuf[p];
    for (int c = tid; c < 128; c += 256) s[p][c] *= pinv;
    __syncthreads();
    for (int idx = tid; idx < 64 * 128; idx += 256) {
      int r = idx >> 7, c = idx & 127;
      if (r != p) s[r][c] -= colbuf[r] * s[p][c];
    }
    __syncthreads();
  }
  for (int idx = tid; idx < 64 * 64; idx += 256) {
    int r = idx >> 6, c = idx & 63;
    invD[idx] = s[r][64 + c];
  }
}

// Snapshot (negated) multiplier column-block before elimination.
__global__ void k_gj_mult(const float* __restrict__ Mgj, float* __restrict__ mult, int pc) {
  int idx = blockIdx.x * blockDim.x + threadIdx.x;
  if (idx >= 512 * 64) return;
  int r = idx >> 6, k = idx & 63;
  mult[idx] = -Mgj[(size_t)r * 1024 + pc + k];
}

// ---------------- recurrence: one 16-column block per launch ----------------
// v = bpre + uu @ D12_blk^T + w[:, :c0] @ D11_blk^T ; then 16 sequential tanh steps.
__global__ void __launch_bounds__(256) k_recur(
    const float* __restrict__ uu, const float* __restrict__ D12,
    const float* __restrict__ D11, const float* __restrict__ rlam,
    const float* __restrict__ bpre, float* __restrict__ w, int kb) {
  __shared__ float vsh[256][17];
  __shared__ float dblk[16][16];
  __shared__ float bsh[16], rsh[16];
  int tid = threadIdx.x;
  int c0 = kb << 4;
  {
    int i = tid >> 4, j = tid & 15;
    dblk[i][j] = D11[(size_t)(c0 + i) * 512 + c0 + j];
    if (tid < 16) { bsh[tid] = bpre[c0 + tid]; rsh[tid] = rlam[c0 + tid]; }
  }
  int lane = tid & 31, wv = tid >> 5;
  int lm = lane & 15, lk = (lane >> 4) << 1;
  int cn = lane & 15, cm = (lane >> 4) << 3;
  int m0 = blockIdx.x * 256 + wv * 32;

  f32x8 acc[2];
#pragma unroll
  for (int t = 0; t < 2; ++t)
#pragma unroll
    for (int v = 0; v < 8; ++v) acc[t][v] = 0.0f;

  // segment 1: uu @ D12_blk^T  (K = 256), software-pipelined
  {
    f32x2 a_c[2], a_n[2], b_c, b_n;
    b_c = *(const f32x2*)(D12 + (size_t)(c0 + lm) * 256 + lk);
#pragma unroll
    for (int t = 0; t < 2; ++t)
      a_c[t] = *(const f32x2*)(uu + (size_t)(m0 + t * 16 + lm) * 256 + lk);
    for (int k = 0; k < 256; k += 4) {
      bool hn = (k + 4) < 256;
      if (hn) {
        b_n = *(const f32x2*)(D12 + (size_t)(c0 + lm) * 256 + k + 4 + lk);
#pragma unroll
        for (int t = 0; t < 2; ++t)
          a_n[t] = *(const f32x2*)(uu + (size_t)(m0 + t * 16 + lm) * 256 + k + 4 + lk);
      }
#pragma unroll
      for (int t = 0; t < 2; ++t) acc[t] = wmma4(a_c[t], b_c, acc[t]);
      if (hn) { b_c = b_n; a_c[0] = a_n[0]; a_c[1] = a_n[1]; }
    }
  }
  // segment 2: w[:, :c0] @ D11_blk^T  (K = c0), software-pipelined
  if (c0 > 0) {
    f32x2 a_c[2], a_n[2], b_c, b_n;
    b_c = *(const f32x2*)(D11 + (size_t)(c0 + lm) * 512 + lk);
#pragma unroll
    for (int t = 0; t < 2; ++t)
      a_c[t] = *(const f32x2*)(w + (size_t)(m0 + t * 16 + lm) * 512 + lk);
    for (int k = 0; k < c0; k += 4) {
      bool hn = (k + 4) < c0;
      if (hn) {
        b_n = *(const f32x2*)(D11 + (size_t)(c0 + lm) * 512 + k + 4 + lk);
#pragma unroll
        for (int t = 0; t < 2; ++t)
          a_n[t] = *(const f32x2*)(w + (size_t)(m0 + t * 16 + lm) * 512 + k + 4 + lk);
      }
#pragma unroll
      for (int t = 0; t < 2; ++t) acc[t] = wmma4(a_c[t], b_c, acc[t]);
      if (hn) { b_c = b_n; a_c[0] = a_n[0]; a_c[1] = a_n[1]; }
    }
  }
#pragma unroll
  for (int t = 0; t < 2; ++t)
#pragma unroll
    for (int v = 0; v < 8; ++v) vsh[wv * 32 + t * 16 + cm + v][cn] = acc[t][v];
  __syncthreads();

  // sequential 16-step tanh recurrence, one batch row per thread
  float wl[16], vv[16];
#pragma unroll
  for (int i = 0; i < 16; ++i) vv[i] = vsh[tid][i] + bsh[i];
#pragma unroll
  for (int i = 0; i < 16; ++i) {
    float vi = vv[i];
    for (int j = 0; j < i; ++j) vi += dblk[i][j] * wl[j];
    wl[i] = tanhf(vi * rsh[i]);
  }
  size_t rbase = (size_t)(blockIdx.x * 256 + tid) * 512 + c0;
#pragma unroll
  for (int i = 0; i < 16; ++i) w[rbase + i] = wl[i];
}

// ---------------- final batch GEMM: y = w@W1^T + uu@W2^T + bias_y ----------------
__global__ void __launch_bounds__(256) k_out(
    const float* __restrict__ w, const float* __restrict__ uu,
    const float* __restrict__ W1, const float* __restrict__ W2,
    const float* __restrict__ biasy, float* __restrict__ y) {
  int tid = threadIdx.x;
  int lane = tid & 31, wv = tid >> 5;
  int lm = lane & 15, lk = (lane >> 4) << 1;
  int cn = lane & 15, cm = (lane >> 4) << 3;
  int rowBlk = blockIdx.x >> 2, colBlk = blockIdx.x & 3;
  int m0 = rowBlk * 256 + wv * 32;
  int n0 = colBlk * 64;

  f32x8 acc[2][4];
#pragma unroll
  for (int t = 0; t < 2; ++t)
#pragma unroll
    for (int u = 0; u < 4; ++u)
#pragma unroll
      for (int v = 0; v < 8; ++v) acc[t][u][v] = 0.0f;

  f32x2 a_c[2], b_c[4], a_n[2], b_n[4];
  // preload k=0 of segment 1 (w / W1)
#pragma unroll
  for (int u = 0; u < 4; ++u)
    b_c[u] = *(const f32x2*)(W1 + (size_t)(n0 + u * 16 + lm) * 512 + lk);
#pragma unroll
  for (int t = 0; t < 2; ++t)
    a_c[t] = *(const f32x2*)(w + (size_t)(m0 + t * 16 + lm) * 512 + lk);

  // segment 1: K = 512 over w/W1; last step prefetches segment 2's first frags
  for (int k = 0; k < 512; k += 4) {
    bool hn = (k + 4) < 512;
    if (hn) {
#pragma unroll
      for (int u = 0; u < 4; ++u)
        b_n[u] = *(const f32x2*)(W1 + (size_t)(n0 + u * 16 + lm) * 512 + k + 4 + lk);
#pragma unroll
      for (int t = 0; t < 2; ++t)
        a_n[t] = *(const f32x2*)(w + (size_t)(m0 + t * 16 + lm) * 512 + k + 4 + lk);
    } else {  // cross-segment prefetch: uu / W2 at k=0
#pragma unroll
      for (int u = 0; u < 4; ++u)
        b_n[u] = *(const f32x2*)(W2 + (size_t)(n0 + u * 16 + lm) * 256 + lk);
#pragma unroll
      for (int t = 0; t < 2; ++t)
        a_n[t] = *(const f32x2*)(uu + (size_t)(m0 + t * 16 + lm) * 256 + lk);
    }
#pragma unroll
    for (int t = 0; t < 2; ++t)
#pragma unroll
      for (int u = 0; u < 4; ++u) acc[t][u] = wmma4(a_c[t], b_c[u], acc[t][u]);
#pragma unroll
    for (int t = 0; t < 2; ++t) a_c[t] = a_n[t];
#pragma unroll
    for (int u = 0; u < 4; ++u) b_c[u] = b_n[u];
  }
  // segment 2: K = 256 over uu/W2
  for (int k = 0; k < 256; k += 4) {
    bool hn = (k + 4) < 256;
    if (hn) {
#pragma unroll
      for (int u = 0; u < 4; ++u)
        b_n[u] = *(const f32x2*)(W2 + (size_t)(n0 + u * 16 + lm) * 256 + k + 4 + lk);
#pragma unroll
      for (int t = 0; t < 2; ++t)
        a_n[t] = *(const f32x2*)(uu + (size_t)(m0 + t * 16 + lm) * 256 + k + 4 + lk);
    }
#pragma unroll
    for (int t = 0; t < 2; ++t)
#pragma unroll
      for (int u = 0; u < 4; ++u) acc[t][u] = wmma4(a_c[t], b_c[u], acc[t][u]);
    if (hn) {
#pragma unroll
      for (int t = 0; t < 2; ++t) a_c[t] = a_n[t];
#pragma unroll
      for (int u = 0; u < 4; ++u) b_c[u] = b_n[u];
    }
  }
#pragma unroll
  for (int t = 0; t < 2; ++t)
#pragma unroll
    for (int u = 0; u < 4; ++u)
#pragma unroll
      for (int v = 0; v < 8; ++v) {
        int r = m0 + t * 16 + cm + v, col = n0 + u * 16 + cn;
        y[(size_t)r * 256 + col] = acc[t][u][v] + biasy[col];
      }
}

// ---------------- host orchestration ----------------
extern "C" void kernel_launch(void* const* d_in, const int* in_sizes, int n_in,
                              void* d_out, int out_size, void* d_ws, size_t ws_size,
                              hipStream_t stream) {
  const float* u_in = (const float*)d_in[0];
  const float* X    = (const float*)d_in[1];
  const float* Y    = (const float*)d_in[2];
  const float* B2   = (const float*)d_in[3];
  const float* C2   = (const float*)d_in[4];
  const float* D21  = (const float*)d_in[5];
  const float* D22  = (const float*)d_in[6];
  const float* D12  = (const float*)d_in[7];
  const float* x0   = (const float*)d_in[8];
  float* yout = (float*)d_out;

  float* ws = (float*)d_ws;
  size_t off = 0;
  auto alloc = [&](size_t n) { float* p = ws + off; off += n; return p; };
  float* H      = alloc(1536ull * 1536);
  float* Mgj    = alloc(512ull * 1024);   // [E | I] -> [I | invE]
  float* D11    = alloc(512ull * 512);
  float* rlam   = alloc(512);
  float* bpre   = alloc(512);
  float* bz     = alloc(512);
  float* invD   = alloc(64 * 64);
  float* mult   = alloc(512 * 64);
  float* rowbuf = alloc(64 * 1024);
  float* G      = alloc(256ull * 512);
  float* W1     = alloc(256ull * 512);
  float* W2     = alloc(256ull * 256);
  float* biasy  = alloc(256);
  float* wbuf   = alloc((size_t)32768 * 512);

  // 1) H = X^T X + eps*I   (A stored transposed: AT=true)
  {
    int tiles = (1536 / 16) * (1536 / 16);
    k_gemm<true, false><<<(tiles + 7) / 8, 256, 0, stream>>>(
        H, 1536, X, 1536, X, 1536, nullptr, 0, 1536, 1536, 1536, EPS_F, -1, -1);
  }
  // 2) derived params
  k_build_EM<<<(512 * 512) / 256, 256, 0, stream>>>(H, Y, Mgj, D11);
  k_vecs<<<2, 256, 0, stream>>>(H, x0, rlam, bpre, bz);

  // 3) blocked Gauss-Jordan inversion of E (8 x 64-wide block pivots)
  for (int it = 0; it < 8; ++it) {
    int pk = it * 64;
    k_gj_diag<<<1, 256, 0, stream>>>(Mgj, invD, pk);
    {  // pivot rows: rowbuf = invD @ M[pk:pk+64, :]
      int tiles = (64 / 16) * (1024 / 16);
      k_gemm<false, false><<<tiles / 8, 256, 0, stream>>>(
          rowbuf, 1024, invD, 64, Mgj + (size_t)pk * 1024, 1024, nullptr, 0,
          64, 1024, 64, 0.f, -1, -1);
    }
    hipMemcpyAsync(Mgj + (size_t)pk * 1024, rowbuf, 64 * 1024 * sizeof(float),
                   hipMemcpyDeviceToDevice, stream);
    k_gj_mult<<<(512 * 64) / 256, 256, 0, stream>>>(Mgj, mult, pk);
    {  // eliminate: M[r,:] += mult[r,:] @ M[pk:pk+64, :] for r outside pivot block
      int tiles = (512 / 16) * (1024 / 16);
      k_gemm<false, false><<<tiles / 8, 256, 0, stream>>>(
          Mgj, 1024, mult, 64, Mgj + (size_t)pk * 1024, 1024, Mgj, 1024,
          512, 1024, 64, 0.f, pk, pk + 64);
    }
  }

  // 4) fold output path: G = C2 @ invE ; W1 = G@B1 + D21 ; W2 = G@B2 + D22
  {
    int tiles = (256 / 16) * (512 / 16);
    k_gemm<false, false><<<tiles / 8, 256, 0, stream>>>(
        G, 512, C2, 512, Mgj + 512, 1024, nullptr, 0, 256, 512, 512, 0.f, -1, -1);
  }
  {  // B1 = H32 = H[1024+k][512+n]
    int tiles = (256 / 16) * (512 / 16);
    k_gemm<false, false><<<tiles / 8, 256, 0, stream>>>(
        W1, 512, G, 512, H + (size_t)1024 * 1536 + 512, 1536, D21, 512,
        256, 512, 512, 0.f, -1, -1);
  }
  {
    int tiles = (256 / 16) * (256 / 16);
    k_gemm<false, false><<<tiles / 8, 256, 0, stream>>>(
        W2, 256, G, 512, B2, 256, D22, 256, 256, 256, 512, 0.f, -1, -1);
  }
  k_biasy<<<1, 256, 0, stream>>>(G, bz, biasy);

  // 5) triangular tanh recurrence: 32 sequential 16-column blocks
  for (int kb = 0; kb < 32; ++kb)
    k_recur<<<32768 / 256, 256, 0, stream>>>(u_in, D12, D11, rlam, bpre, wbuf, kb);

  // 6) y = w @ W1^T + uu @ W2^T + bias_y
  k_out<<<(32768 / 256) * 4, 256, 0, stream>>>(wbuf, u_in, W1, W2, biasy, yout);
}